// PosFeatureLayer_83416854823346
// MI455X (gfx1250) — compile-verified
//
#include <hip/hip_runtime.h>
#include <hip/hip_bf16.h>

typedef __attribute__((ext_vector_type(2))) float v2f;
typedef __attribute__((ext_vector_type(4))) float v4f;
typedef __attribute__((ext_vector_type(8))) float v8f;

#define B_SZ 16
#define M_SZ 8192
#define N_SZ 8192
#define D_SZ 256

// One wave = one 16-row x 256-col output slab.
// out = emb + mask * (gathered_feat[16x4] @ W[4x256]) via 16x V_WMMA_F32_16X16X4_F32
__global__ __launch_bounds__(256) void PosFeatureLayer_kernel(
    const float* __restrict__ emb,
    const int* __restrict__ num,
    const float* __restrict__ pts,
    const int* __restrict__ idxs,
    const long long* __restrict__ image_shape,
    const float* __restrict__ Wg,
    float* __restrict__ out)
{
    __shared__ float ldsW[4 * D_SZ];  // 4KB: all of W

    const int tid = threadIdx.x;

    // Stage W into LDS: 256 threads x float4 = 4KB
    {
        const v4f* src = (const v4f*)Wg;
        v4f* dst = (v4f*)ldsW;
        dst[tid] = src[tid];
    }
    __syncthreads();

    const int wave = (blockIdx.x << 3) + (tid >> 5);   // 8 waves/block
    const int lane = tid & 31;
    const int b  = wave >> 9;                          // / (M/16)
    const int m0 = (wave & 511) << 4;

    // --- normalization constants (from device-resident image_shape int64[4]) ---
    const float Hf = (float)image_shape[2];
    const float Wf = (float)image_shape[3];
    const float cx = Wf * 0.5f;
    const float cy = Hf * 0.5f;
    const float inv_s = 1.0f / (fmaxf(Wf, Hf) * 0.7f);
    const float lsc   = sqrtf(Wf * Wf + Hf * Hf) * 0.7f;
    const float inv_l = 1.0f / lsc;

    // --- A fragment (16x4 f32): lanes 0-15 -> K=0,1 ; lanes 16-31 -> K=2,3 ---
    const int n15 = lane & 15;        // row M (for A) / col N (for B,C,D)
    const int hi  = lane >> 4;
    const int r   = m0 + n15;         // output row this lane's A-slice belongs to
    const int nb  = num[b];
    const bool active = (r < nb);
    const int idx = idxs[b * M_SZ + r];
    const float* p = pts + ((size_t)b * N_SZ + (size_t)idx) * 5;
    const float f0 = (p[0] - cx) * inv_s;                 // upts.x
    const float f1 = (p[1] - cy) * inv_s;                 // upts.y
    const float f2 = (p[4] - 45.0f) * (1.0f / 126.0f);    // uangle (180*0.7)
    const float f3 = (p[3] - lsc * 0.5f) * inv_l;         // ulen

    v2f a;
    a.x = active ? (hi ? f2 : f0) : 0.0f;   // mask -> zero contribution (jnp.where)
    a.y = active ? (hi ? f3 : f1) : 0.0f;

    // C/D layout: VGPR j holds M = m0 + j + 8*hi, N = n15
    const size_t rowbase = ((size_t)b * M_SZ + (size_t)(m0 + (hi << 3))) * D_SZ + n15;
    // B layout: lanes 0-15 -> K=0,1 ; lanes 16-31 -> K=2,3 at column n15
    const float* wlds = ldsW + ((hi << 1) * D_SZ) + n15;

    #pragma unroll
    for (int t = 0; t < 16; ++t) {
        const int d0 = t << 4;

        v2f bf;
        bf.x = wlds[d0];
        bf.y = wlds[d0 + D_SZ];

        v8f c;
        #pragma unroll
        for (int j = 0; j < 8; ++j)
            c[j] = __builtin_nontemporal_load(&emb[rowbase + (size_t)j * D_SZ + d0]);

        // D = A(16x4) x B(4x16) + C   -- emb add fused into the accumulator
        v8f d = __builtin_amdgcn_wmma_f32_16x16x4_f32(
            /*neg_a=*/false, a, /*neg_b=*/false, bf,
            /*c_mod=*/(short)0, c, /*reuse_a=*/false, /*reuse_b=*/false);

        #pragma unroll
        for (int j = 0; j < 8; ++j)
            __builtin_nontemporal_store(d[j], &out[rowbase + (size_t)j * D_SZ + d0]);
    }
}

extern "C" void kernel_launch(void* const* d_in, const int* in_sizes, int n_in,
                              void* d_out, int out_size, void* d_ws, size_t ws_size,
                              hipStream_t stream) {
    const float*     emb = (const float*)d_in[0];
    const int*       num = (const int*)d_in[1];
    const float*     pts = (const float*)d_in[2];
    const int*       idx = (const int*)d_in[3];
    const long long* shp = (const long long*)d_in[4];
    const float*     W   = (const float*)d_in[5];
    float* out = (float*)d_out;

    // B*M/16 tiles = 8192 waves; 8 waves per 256-thread block -> 1024 blocks
    dim3 grid(B_SZ * M_SZ / 16 / 8);
    dim3 block(256);
    PosFeatureLayer_kernel<<<grid, block, 0, stream>>>(emb, num, pts, idx, shp, W, out);
}